// SE3Transformer_35399120454064
// MI455X (gfx1250) — compile-verified
//
#include <hip/hip_runtime.h>

typedef float v2f __attribute__((ext_vector_type(2)));
typedef float v8f __attribute__((ext_vector_type(8)));

struct RadialP { const float *w1,*b1,*g1,*bb1,*w2,*b2,*g2,*bb2,*w3,*b3; };

__device__ __forceinline__ v8f wmma4(v2f a, v2f b, v8f c) {
  // D[16x16] += A[16x4] * B[4x16], all f32 (CDNA5 V_WMMA_F32_16X16X4_F32)
  return __builtin_amdgcn_wmma_f32_16x16x4_f32(false, a, false, b, (short)0, c, false, false);
}

// XOR-butterfly lane exchange via ds_swizzle group-of-32 mode:
// new_lane = ((lane & 0x1f) | 0) ^ M  -> pattern = (M<<10) | 0x1f.
// Masks 1/2/4/8 keep the two 16-lane halves independent (wave32).
template<int M>
__device__ __forceinline__ float se3_swz_xor(float x) {
  return __int_as_float(__builtin_amdgcn_ds_swizzle(__float_as_int(x), (M << 10) | 0x1f));
}

// LayerNorm(+bias)+ReLU over 32 channels held as two 16x16 C/D tiles.
// Row M's 32 values live in {u0[j],u1[j]} across one 16-lane half -> butterfly reduce.
__device__ __forceinline__ void ln_relu32(v8f& u0, v8f& u1,
    const float* __restrict__ bias, const float* __restrict__ g,
    const float* __restrict__ bb, int lm)
{
  float biasA = bias[lm],  biasB = bias[16+lm];
  float gA = g[lm],        gB = g[16+lm];
  float bbA = bb[lm],      bbB = bb[16+lm];
#pragma unroll
  for (int j = 0; j < 8; ++j) {
    float x0 = u0[j] + biasA, x1 = u1[j] + biasB;
    float s = x0 + x1, q = x0*x0 + x1*x1;
    s += se3_swz_xor<1>(s);  q += se3_swz_xor<1>(q);
    s += se3_swz_xor<2>(s);  q += se3_swz_xor<2>(q);
    s += se3_swz_xor<4>(s);  q += se3_swz_xor<4>(q);
    s += se3_swz_xor<8>(s);  q += se3_swz_xor<8>(q);
    float mu  = s * (1.0f/32.0f);
    float var = fmaxf(q * (1.0f/32.0f) - mu*mu, 0.0f);
    float rs  = rsqrtf(var + 1e-5f);
    u0[j] = fmaxf((x0 - mu) * rs * gA + bbA, 0.0f);
    u1[j] = fmaxf((x1 - mu) * rs * gB + bbB, 0.0f);
  }
}

__global__ void se3_fill(float* __restrict__ p, float v, int n) {
  int i = blockIdx.x * blockDim.x + threadIdx.x;
  if (i < n) p[i] = v;
}

__global__ void se3_edge_geom(const float* __restrict__ pos, const float* __restrict__ wfe,
                              const int* __restrict__ src, const int* __restrict__ dst,
                              float* __restrict__ dhat, float* __restrict__ feat8, int E)
{
  int e = blockIdx.x * blockDim.x + threadIdx.x;
  if (e >= E) return;
  int s = src[e], d = dst[e];
  float dx = pos[d*3+0] - pos[s*3+0];
  float dy = pos[d*3+1] - pos[s*3+1];
  float dz = pos[d*3+2] - pos[s*3+2];
  float r  = sqrtf(dx*dx + dy*dy + dz*dz);
  float iv = 1.0f / fmaxf(r, 1e-6f);
  dhat[(size_t)e*3+0] = dx*iv; dhat[(size_t)e*3+1] = dy*iv; dhat[(size_t)e*3+2] = dz*iv;
  float* f8 = feat8 + (size_t)e*8;
  f8[0] = wfe[(size_t)e*4+0]; f8[1] = wfe[(size_t)e*4+1];
  f8[2] = wfe[(size_t)e*4+2]; f8[3] = wfe[(size_t)e*4+3];
  f8[4] = r; f8[5] = 0.f; f8[6] = 0.f; f8[7] = 0.f;   // K padded to 8
}

// Fused radial MLP (feat[E,5] -> 32 -> 32 -> K3) + equivariant edge-conv contraction.
// One wave per 16-edge tile; WMMA f32 16x16x4 for all three GEMMs.
// IN_DEG: 0 -> contract with h0[src]; 1 -> contract with s = h1[src].dhat (and h1 basis for NF=3)
// OUT_DEG 0/NF=1: acc0[e,o]  += sum_c R[o,c]*x[c]
// OUT_DEG 1/NF=1: acc1[e,o,:] += (sum_c R[o,c]*x[c]) * dhat
// OUT_DEG 1/NF=3: acc1[e,o,:] += sum_c R0*h1 + R1*(dhat x h1) + R2*(dhat*s - h1/3)
template<int CO, int NF, int IN_DEG, int OUT_DEG, bool DST_ATOMIC>
__global__ __launch_bounds__(128)
void se3_radial_conv(const float* __restrict__ feat8, const float* __restrict__ dhat,
                     const int* __restrict__ src, const int* __restrict__ dstv,
                     const float* __restrict__ h0, const float* __restrict__ h1,
                     RadialP rp, float* __restrict__ out, int E)
{
  constexpr int K3 = CO * 16 * NF;
  constexpr int NT = K3 / 16;
  __shared__ float s_w1[8*32];
  __shared__ float s_w2[32*32];
  __shared__ float s_act[4][16*36];
  __shared__ float s_r[4][16*17];
  __shared__ float s_x[4][16*16];
  __shared__ float s_h1[4][16*48];
  __shared__ float s_dh[4][16*3];

  const int t    = threadIdx.x;
  const int lane = t & 31;
  const int w    = t >> 5;
  const int hlf  = lane >> 4;      // 0 or 1 (half-wave)
  const int lm   = lane & 15;
  const int kb   = hlf * 2;        // A-layout K base: lanes 0-15 -> K0,K1 ; 16-31 -> K2,K3
  const int base = (blockIdx.x * 4 + w) * 16;

  // cooperative weight staging (w1 zero-padded K=5..7)
  for (int i = t; i < 8*32; i += 128) {
    int kk = i >> 5, cc = i & 31;
    s_w1[i] = (kk < 5) ? rp.w1[kk*32 + cc] : 0.0f;
  }
  for (int i = t; i < 32*32; i += 128) s_w2[i] = rp.w2[i];

  if (lane < 16) {
#pragma unroll
    for (int a = 0; a < 3; ++a) s_dh[w][lane*3+a] = dhat[(size_t)(base+lane)*3 + a];
  }
  if constexpr (IN_DEG == 0) {
    for (int i = lane; i < 256; i += 32) {
      int e = i >> 4, c = i & 15;
      s_x[w][i] = h0[(size_t)src[base+e]*16 + c];
    }
  } else {
    for (int i = lane; i < 768; i += 32) {
      int e = i / 48, q = i - e*48;
      s_h1[w][i] = h1[(size_t)src[base+e]*48 + q];
    }
  }
  __syncthreads();
  if constexpr (IN_DEG == 1) {
    for (int i = lane; i < 256; i += 32) {   // s[e,c] = h1[e,c,:].dhat[e,:]
      int e = i >> 4, c = i & 15;
      s_x[w][i] = s_h1[w][e*48+c*3+0]*s_dh[w][e*3+0]
                + s_h1[w][e*48+c*3+1]*s_dh[w][e*3+1]
                + s_h1[w][e*48+c*3+2]*s_dh[w][e*3+2];
    }
    __syncthreads();
  }

  __builtin_prefetch(rp.w3 + (size_t)lane*16, 0, 0);

  // ---- stage 1: feat[16x8] @ w1p[8x32] ----
  v8f y0 = {0,0,0,0,0,0,0,0}, y1 = {0,0,0,0,0,0,0,0};
#pragma unroll
  for (int ks = 0; ks < 2; ++ks) {
    int kk = ks*4 + kb;
    v2f a; const float* fp = feat8 + (size_t)(base+lm)*8 + kk;
    a.x = fp[0]; a.y = fp[1];
    v2f bA; bA.x = s_w1[kk*32+lm];    bA.y = s_w1[(kk+1)*32+lm];
    v2f bB; bB.x = s_w1[kk*32+16+lm]; bB.y = s_w1[(kk+1)*32+16+lm];
    y0 = wmma4(a, bA, y0);
    y1 = wmma4(a, bB, y1);
  }
  ln_relu32(y0, y1, rp.b1, rp.g1, rp.bb1, lm);
  __syncthreads();
#pragma unroll
  for (int j = 0; j < 8; ++j) {
    int row = j + hlf*8;
    s_act[w][row*36+lm]    = y0[j];
    s_act[w][row*36+16+lm] = y1[j];
  }
  __syncthreads();

  // ---- stage 2: A1[16x32] @ w2[32x32] ----
  v8f z0 = {0,0,0,0,0,0,0,0}, z1 = {0,0,0,0,0,0,0,0};
#pragma unroll
  for (int ks = 0; ks < 8; ++ks) {
    int kk = ks*4 + kb;
    v2f a; a.x = s_act[w][lm*36+kk]; a.y = s_act[w][lm*36+kk+1];
    v2f bA; bA.x = s_w2[kk*32+lm];    bA.y = s_w2[(kk+1)*32+lm];
    v2f bB; bB.x = s_w2[kk*32+16+lm]; bB.y = s_w2[(kk+1)*32+16+lm];
    z0 = wmma4(a, bA, z0);
    z1 = wmma4(a, bB, z1);
  }
  ln_relu32(z0, z1, rp.b2, rp.g2, rp.bb2, lm);
  __syncthreads();
#pragma unroll
  for (int j = 0; j < 8; ++j) {
    int row = j + hlf*8;
    s_act[w][row*36+lm]    = z0[j];
    s_act[w][row*36+16+lm] = z1[j];
  }
  __syncthreads();

  // ---- stage 3: A2[16x32] @ w3[32xK3], per 16-col n-tile + fused contraction ----
  for (int nt = 0; nt < NT; ++nt) {
    v8f r = {0,0,0,0,0,0,0,0};
#pragma unroll
    for (int ks = 0; ks < 8; ++ks) {
      int kk = ks*4 + kb;
      v2f a; a.x = s_act[w][lm*36+kk]; a.y = s_act[w][lm*36+kk+1];
      v2f b; b.x = rp.w3[(size_t)kk*K3 + nt*16 + lm];
             b.y = rp.w3[(size_t)(kk+1)*K3 + nt*16 + lm];
      r = wmma4(a, b, r);
    }
    float b3v = rp.b3[nt*16 + lm];
#pragma unroll
    for (int j = 0; j < 8; ++j) {
      int row = j + hlf*8;
      s_r[w][row*17+lm] = r[j] + b3v;
    }
    __syncthreads();
    if (lane < 16) {
      int e = lane;
      if constexpr (OUT_DEG == 0) {          // col = o*16 + c, n-tile == o
        float ssum = 0.f;
#pragma unroll
        for (int c = 0; c < 16; ++c) ssum += s_r[w][e*17+c] * s_x[w][e*16+c];
        if constexpr (DST_ATOMIC) atomicAdd(out + (size_t)dstv[base+e]*CO + nt, ssum);
        else                      out[(size_t)(base+e)*CO + nt] += ssum;
      } else if constexpr (NF == 1) {        // '0,1': t[o]*dhat
        float ssum = 0.f;
#pragma unroll
        for (int c = 0; c < 16; ++c) ssum += s_r[w][e*17+c] * s_x[w][e*16+c];
        float* op = out + ((size_t)(base+e)*CO + nt)*3;
        op[0] += ssum * s_dh[w][e*3+0];
        op[1] += ssum * s_dh[w][e*3+1];
        op[2] += ssum * s_dh[w][e*3+2];
      } else {                               // '1,1': col = (o*16+c)*3 + basis
        int o = nt / 3, sub = nt - o*3;
        float dx = s_dh[w][e*3+0], dy = s_dh[w][e*3+1], dz = s_dh[w][e*3+2];
        float a0 = 0.f, a1 = 0.f, a2 = 0.f;
#pragma unroll
        for (int q = 0; q < 16; ++q) {
          int idx = sub*16 + q;
          int c = idx / 3, bsel = idx - c*3;
          float rv = s_r[w][e*17+q];
          float hx = s_h1[w][e*48+c*3+0], hy = s_h1[w][e*48+c*3+1], hz = s_h1[w][e*48+c*3+2];
          float Bx, By, Bz;
          if (bsel == 0)      { Bx = hx; By = hy; Bz = hz; }
          else if (bsel == 1) { Bx = dy*hz - dz*hy; By = dz*hx - dx*hz; Bz = dx*hy - dy*hx; }
          else {
            float sv = s_x[w][e*16+c];
            Bx = dx*sv - hx*(1.f/3.f); By = dy*sv - hy*(1.f/3.f); Bz = dz*sv - hz*(1.f/3.f);
          }
          a0 += rv*Bx; a1 += rv*By; a2 += rv*Bz;
        }
        float* op = out + ((size_t)(base+e)*CO + o)*3;
        op[0] += a0; op[1] += a1; op[2] += a2;
      }
    }
    __syncthreads();
  }
}

__global__ void se3_qd(const float* __restrict__ q0, const float* __restrict__ q1,
                       const float* __restrict__ h0, const float* __restrict__ h1,
                       float* __restrict__ qd0, float* __restrict__ qd1, int N, int hasdeg1)
{
  int n = blockIdx.x * blockDim.x + threadIdx.x;
  if (n >= N) return;
  for (int o = 0; o < 4; ++o) {
    float s = 0.f;
    for (int c = 0; c < 16; ++c) s += q0[o*16+c] * h0[(size_t)n*16+c];
    qd0[(size_t)n*4+o] = s;
  }
  if (hasdeg1) {
    for (int o = 0; o < 4; ++o)
      for (int a = 0; a < 3; ++a) {
        float s = 0.f;
        for (int c = 0; c < 16; ++c) s += q1[o*16+c] * h1[((size_t)n*16+c)*3+a];
        qd1[((size_t)n*4+o)*3+a] = s;
      }
  }
}

__device__ __forceinline__ void se3_atomicMaxF(float* addr, float v) {
  unsigned* ua = (unsigned*)addr;
  unsigned cur = __float_as_uint(*addr);
  while (__uint_as_float(cur) < v) {
    unsigned old = atomicCAS(ua, cur, __float_as_uint(v));
    if (old == cur) break;
    cur = old;
  }
}

__global__ void se3_logits(const float* __restrict__ k0, const float* __restrict__ k1,
                           const float* __restrict__ qd0, const float* __restrict__ qd1,
                           const int* __restrict__ dst, float* __restrict__ lg,
                           float* __restrict__ mx, float scale, int E, int hasdeg1)
{
  int e = blockIdx.x * blockDim.x + threadIdx.x;
  if (e >= E) return;
  int d = dst[e];
  float s = 0.f;
  for (int o = 0; o < 4; ++o) s += k0[(size_t)e*4+o] * qd0[(size_t)d*4+o];
  if (hasdeg1) {
    for (int i = 0; i < 12; ++i) s += k1[(size_t)e*12+i] * qd1[(size_t)d*12+i];
  }
  s *= scale;
  lg[e] = s;
  se3_atomicMaxF(mx + d, s);
}

__global__ void se3_exp_den(float* __restrict__ lg, const float* __restrict__ mx,
                            float* __restrict__ den, const int* __restrict__ dst, int E)
{
  int e = blockIdx.x * blockDim.x + threadIdx.x;
  if (e >= E) return;
  int d = dst[e];
  float ex = expf(lg[e] - mx[d]);
  lg[e] = ex;
  atomicAdd(den + d, ex);
}

__global__ void se3_aggregate(const float* __restrict__ ex, const float* __restrict__ den,
                              const int* __restrict__ dst, const float* __restrict__ v0,
                              const float* __restrict__ v1, float* __restrict__ agg0,
                              float* __restrict__ agg1, int E)
{
  int e = blockIdx.x * blockDim.x + threadIdx.x;
  if (e >= E) return;
  int d = dst[e];
  float al = ex[e] / fmaxf(den[d], 1e-9f);
  for (int o = 0; o < 4; ++o)  atomicAdd(agg0 + (size_t)d*4 + o,  al * v0[(size_t)e*4 + o]);
  for (int i = 0; i < 12; ++i) atomicAdd(agg1 + (size_t)d*12 + i, al * v1[(size_t)e*12 + i]);
}

__global__ void se3_deg(const int* __restrict__ dst, float* __restrict__ deg, int E) {
  int e = blockIdx.x * blockDim.x + threadIdx.x;
  if (e < E) atomicAdd(deg + dst[e], 1.0f);
}

__global__ void se3_proj_norm(const float* __restrict__ agg0, const float* __restrict__ agg1,
                              const float* __restrict__ h0in, const float* __restrict__ h1in,
                              const float* __restrict__ proj0, const float* __restrict__ proj1,
                              const float* __restrict__ g0, const float* __restrict__ b0,
                              const float* __restrict__ g1, const float* __restrict__ b1,
                              float* __restrict__ h0out, float* __restrict__ h1out,
                              int N, int has_h1)
{
  int n = blockIdx.x * blockDim.x + threadIdx.x;
  if (n >= N) return;
  float cat0[20];
  for (int o = 0; o < 4; ++o)  cat0[o]   = agg0[(size_t)n*4+o];
  for (int c = 0; c < 16; ++c) cat0[4+c] = h0in[(size_t)n*16+c];
  float t0[16];
  for (int o = 0; o < 16; ++o) {
    float s = 0.f;
    for (int c = 0; c < 20; ++c) s += proj0[o*20+c] * cat0[c];
    t0[o] = s;
  }
  float t1[16][3];
  if (has_h1) {
    float cat1[20][3];
    for (int o = 0; o < 4; ++o)  for (int a = 0; a < 3; ++a) cat1[o][a]   = agg1[((size_t)n*4+o)*3+a];
    for (int c = 0; c < 16; ++c) for (int a = 0; a < 3; ++a) cat1[4+c][a] = h1in[((size_t)n*16+c)*3+a];
    for (int o = 0; o < 16; ++o)
      for (int a = 0; a < 3; ++a) {
        float s = 0.f;
        for (int c = 0; c < 20; ++c) s += proj1[o*20+c] * cat1[c][a];
        t1[o][a] = s;
      }
  } else {
    for (int o = 0; o < 16; ++o)
      for (int a = 0; a < 3; ++a) {
        float s = 0.f;
        for (int c = 0; c < 4; ++c) s += proj1[o*4+c] * agg1[((size_t)n*4+c)*3+a];
        t1[o][a] = s;
      }
  }
  { // gnorm degree 0
    float nr[16]; float mu = 0.f;
    for (int c = 0; c < 16; ++c) { nr[c] = sqrtf(t0[c]*t0[c] + 1e-12f); mu += nr[c]; }
    mu *= (1.f/16.f);
    float var = 0.f;
    for (int c = 0; c < 16; ++c) { float d = nr[c]-mu; var += d*d; }
    var *= (1.f/16.f);
    float rs = rsqrtf(var + 1e-5f);
    for (int c = 0; c < 16; ++c) {
      float tt = fmaxf((nr[c]-mu)*rs*g0[c] + b0[c], 0.f);
      h0out[(size_t)n*16+c] = tt * (t0[c] / nr[c]);
    }
  }
  { // gnorm degree 1
    float nr[16]; float mu = 0.f;
    for (int c = 0; c < 16; ++c) {
      nr[c] = sqrtf(t1[c][0]*t1[c][0] + t1[c][1]*t1[c][1] + t1[c][2]*t1[c][2] + 1e-12f);
      mu += nr[c];
    }
    mu *= (1.f/16.f);
    float var = 0.f;
    for (int c = 0; c < 16; ++c) { float d = nr[c]-mu; var += d*d; }
    var *= (1.f/16.f);
    float rs = rsqrtf(var + 1e-5f);
    for (int c = 0; c < 16; ++c) {
      float tt = fmaxf((nr[c]-mu)*rs*g1[c] + b1[c], 0.f);
      float iv = tt / nr[c];
      for (int a = 0; a < 3; ++a) h1out[((size_t)n*16+c)*3+a] = t1[c][a] * iv;
    }
  }
}

__global__ void se3_final_node(const float* __restrict__ nodeacc, const float* __restrict__ deg,
                               const float* __restrict__ h0, const float* __restrict__ selfw,
                               float* __restrict__ gm, int N)
{
  int n = blockIdx.x * blockDim.x + threadIdx.x;
  if (n >= N) return;
  float iv = 1.0f / fmaxf(deg[n], 1.0f);
  for (int o = 0; o < 32; ++o) {
    float s = 0.f;
    for (int c = 0; c < 16; ++c) s += selfw[o*16+c] * h0[(size_t)n*16+c];
    float hf = nodeacc[(size_t)n*32+o] * iv + s;
    atomicAdd(gm + (size_t)(n >> 7)*32 + o, hf * (1.0f/128.0f));   // NPG=128 mean pooling
  }
}

__global__ void se3_head(const float* __restrict__ gm, const float* __restrict__ fc1w,
                         const float* __restrict__ fc1b, const float* __restrict__ fc2w,
                         const float* __restrict__ fc2b, float* __restrict__ out, int G)
{
  int g = blockIdx.x * blockDim.x + threadIdx.x;
  if (g >= G) return;
  float x[32];
  for (int i = 0; i < 32; ++i) x[i] = gm[(size_t)g*32+i];
  float acc = 0.f;
  for (int j = 0; j < 32; ++j) {
    float s = fc1b[j];
    for (int i = 0; i < 32; ++i) s += x[i] * fc1w[i*32+j];
    acc += fmaxf(s, 0.f) * fc2w[j];
  }
  out[g] = acc + fc2b[0];
}

// ---- host side ----
static inline const float* Pf(void* const* d_in, int i) { return (const float*)d_in[i]; }

static RadialP mk_rp(void* const* d_in, int b) {
  // jax tree_flatten (sorted keys): b1,b2,b3,bb1,bb2,g1,g2,w1,w2,w3
  RadialP r;
  r.b1  = Pf(d_in, b+0); r.b2  = Pf(d_in, b+1); r.b3 = Pf(d_in, b+2);
  r.bb1 = Pf(d_in, b+3); r.bb2 = Pf(d_in, b+4);
  r.g1  = Pf(d_in, b+5); r.g2  = Pf(d_in, b+6);
  r.w1  = Pf(d_in, b+7); r.w2  = Pf(d_in, b+8); r.w3 = Pf(d_in, b+9);
  return r;
}

extern "C" void kernel_launch(void* const* d_in, const int* in_sizes, int n_in,
                              void* d_out, int out_size, void* d_ws, size_t ws_size,
                              hipStream_t stream)
{
  (void)n_in; (void)ws_size;
  const float* POS = Pf(d_in, 0);
  const float* F   = Pf(d_in, 1);
  const float* WE  = Pf(d_in, 2);
  const int*   SRC = (const int*)d_in[3];
  const int*   DST = (const int*)d_in[4];
  const int E = in_sizes[3];
  const int N = in_sizes[5];
  const int G = out_size;

  float* ws = (float*)d_ws;
  size_t off = 0;
  auto alloc = [&](size_t n) { float* p = ws + off; off += n; return p; };
  float* dhat  = alloc((size_t)3*E);
  float* feat8 = alloc((size_t)8*E);
  float* v0    = alloc((size_t)4*E);
  float* v1    = alloc((size_t)12*E);
  float* kk0   = alloc((size_t)4*E);
  float* kk1   = alloc((size_t)12*E);
  float* lg    = alloc((size_t)E);
  float* mx    = alloc((size_t)N);
  float* den   = alloc((size_t)N);
  float* agg0  = alloc((size_t)4*N);
  float* agg1  = alloc((size_t)12*N);
  float* h0b   = alloc((size_t)16*N);
  float* h1b   = alloc((size_t)48*N);
  float* h0c   = alloc((size_t)16*N);
  float* h1c   = alloc((size_t)48*N);
  float* qd0   = alloc((size_t)4*N);
  float* qd1   = alloc((size_t)12*N);
  float* deg   = alloc((size_t)N);
  float* nacc  = alloc((size_t)32*N);
  float* gm    = alloc((size_t)G*32);

  auto fill = [&](float* p, float v, size_t n) {
    se3_fill<<<(unsigned)((n + 255) / 256), 256, 0, stream>>>(p, v, (int)n);
  };
  const int EB = (E + 255) / 256;
  const int NB = (N + 127) / 128;
  const int RG = E / 64;    // radial kernel: 4 waves * 16 edges per block

  se3_edge_geom<<<EB, 256, 0, stream>>>(POS, WE, SRC, DST, dhat, feat8, E);

  // ===== layer 1 (in_degs = (0,)) =====
  fill(kk0, 0.f, (size_t)4*E); fill(v0, 0.f, (size_t)4*E); fill(v1, 0.f, (size_t)12*E);
  fill(mx, -3.0e38f, N); fill(den, 0.f, N);
  fill(agg0, 0.f, (size_t)4*N); fill(agg1, 0.f, (size_t)12*N);
  se3_radial_conv<4,1,0,0,false><<<RG,128,0,stream>>>(feat8, dhat, SRC, DST, F, nullptr, mk_rp(d_in,31), kk0, E);  // k'0,0'
  se3_qd<<<NB,128,0,stream>>>(Pf(d_in,47), nullptr, F, nullptr, qd0, qd1, N, 0);
  se3_logits<<<EB,256,0,stream>>>(kk0, kk1, qd0, qd1, DST, lg, mx, 0.5f, E, 0);      // 1/sqrt(4)
  se3_exp_den<<<EB,256,0,stream>>>(lg, mx, den, DST, E);
  se3_radial_conv<4,1,0,0,false><<<RG,128,0,stream>>>(feat8, dhat, SRC, DST, F, nullptr, mk_rp(d_in,48), v0, E);   // v'0,0'
  se3_radial_conv<4,1,0,1,false><<<RG,128,0,stream>>>(feat8, dhat, SRC, DST, F, nullptr, mk_rp(d_in,58), v1, E);   // v'0,1'
  se3_aggregate<<<EB,256,0,stream>>>(lg, den, DST, v0, v1, agg0, agg1, E);
  se3_proj_norm<<<NB,128,0,stream>>>(agg0, agg1, F, nullptr, Pf(d_in,45), Pf(d_in,46),
                                     Pf(d_in,42), Pf(d_in,41), Pf(d_in,44), Pf(d_in,43),
                                     h0b, h1b, N, 0);

  // ===== layer 2 (in_degs = (0,1)) =====
  fill(kk0, 0.f, (size_t)4*E); fill(kk1, 0.f, (size_t)12*E);
  fill(v0, 0.f, (size_t)4*E);  fill(v1, 0.f, (size_t)12*E);
  fill(mx, -3.0e38f, N); fill(den, 0.f, N);
  fill(agg0, 0.f, (size_t)4*N); fill(agg1, 0.f, (size_t)12*N);
  se3_radial_conv<4,1,0,0,false><<<RG,128,0,stream>>>(feat8, dhat, SRC, DST, h0b, nullptr, mk_rp(d_in,68),  kk0, E); // k'0,0'
  se3_radial_conv<4,1,0,1,false><<<RG,128,0,stream>>>(feat8, dhat, SRC, DST, h0b, nullptr, mk_rp(d_in,78),  kk1, E); // k'0,1'
  se3_radial_conv<4,1,1,0,false><<<RG,128,0,stream>>>(feat8, dhat, SRC, DST, h0b, h1b,     mk_rp(d_in,88),  kk0, E); // k'1,0'
  se3_radial_conv<4,3,1,1,false><<<RG,128,0,stream>>>(feat8, dhat, SRC, DST, h0b, h1b,     mk_rp(d_in,98),  kk1, E); // k'1,1'
  se3_qd<<<NB,128,0,stream>>>(Pf(d_in,114), Pf(d_in,115), h0b, h1b, qd0, qd1, N, 1);
  se3_logits<<<EB,256,0,stream>>>(kk0, kk1, qd0, qd1, DST, lg, mx, 0.25f, E, 1);     // 1/sqrt(16)
  se3_exp_den<<<EB,256,0,stream>>>(lg, mx, den, DST, E);
  se3_radial_conv<4,1,0,0,false><<<RG,128,0,stream>>>(feat8, dhat, SRC, DST, h0b, nullptr, mk_rp(d_in,116), v0, E);  // v'0,0'
  se3_radial_conv<4,1,0,1,false><<<RG,128,0,stream>>>(feat8, dhat, SRC, DST, h0b, nullptr, mk_rp(d_in,126), v1, E);  // v'0,1'
  se3_radial_conv<4,1,1,0,false><<<RG,128,0,stream>>>(feat8, dhat, SRC, DST, h0b, h1b,     mk_rp(d_in,136), v0, E);  // v'1,0'
  se3_radial_conv<4,3,1,1,false><<<RG,128,0,stream>>>(feat8, dhat, SRC, DST, h0b, h1b,     mk_rp(d_in,146), v1, E);  // v'1,1'
  se3_aggregate<<<EB,256,0,stream>>>(lg, den, DST, v0, v1, agg0, agg1, E);
  se3_proj_norm<<<NB,128,0,stream>>>(agg0, agg1, h0b, h1b, Pf(d_in,112), Pf(d_in,113),
                                     Pf(d_in,109), Pf(d_in,108), Pf(d_in,111), Pf(d_in,110),
                                     h0c, h1c, N, 1);

  // ===== final conv + pooling + head =====
  fill(deg, 0.f, N); fill(nacc, 0.f, (size_t)32*N); fill(gm, 0.f, (size_t)G*32);
  se3_deg<<<EB,256,0,stream>>>(DST, deg, E);
  se3_radial_conv<32,1,0,0,true><<<RG,128,0,stream>>>(feat8, dhat, SRC, DST, h0c, nullptr, mk_rp(d_in,10), nacc, E); // final '0,0'
  se3_radial_conv<32,1,1,0,true><<<RG,128,0,stream>>>(feat8, dhat, SRC, DST, h0c, h1c,     mk_rp(d_in,20), nacc, E); // final '1,0'
  se3_final_node<<<NB,128,0,stream>>>(nacc, deg, h0c, Pf(d_in,30), gm, N);
  se3_head<<<1,64,0,stream>>>(gm, Pf(d_in,7), Pf(d_in,6), Pf(d_in,9), Pf(d_in,8), (float*)d_out, G);
}